// CrossStockAttention_24163486007573
// MI455X (gfx1250) — compile-verified
//
#include <hip/hip_runtime.h>
#include <hip/hip_bf16.h>
#include <math.h>

#define B_    4
#define N_    2048
#define D_    256
#define H_    8
#define HD_   32
#define TOPK_ 40
#define LN_EPS 1e-5f
#define SIMLD 2052   // 2048 + 4 pad -> rows land on distinct LDS bank groups

typedef __attribute__((ext_vector_type(2))) float v2f;
typedef __attribute__((ext_vector_type(8))) float v8f;

// ---------------- WMMA helpers (fp32 16x16x4, wave32) ----------------

__device__ __forceinline__ v8f wmma_f32(v2f a, v2f b, v8f c) {
  // (neg_a, A, neg_b, B, c_mod, C, reuse_a, reuse_b)
  return __builtin_amdgcn_wmma_f32_16x16x4_f32(false, a, false, b, (short)0, c,
                                               false, false);
}

// A (16x4, MxK) or B (4x16, KxN) fragment from a ROW-MAJOR source where the
// 16-dim (M or N) is the row index:
//   elem(lane) = src[(lane%16)*stride + kcol + (lane/16)*2 + {0,1}]
__device__ __forceinline__ v2f frag_rm(const float* __restrict__ src, int stride,
                                       int kcol) {
  const int ln = threadIdx.x & 31;
  const float* p = src + (size_t)(ln & 15) * stride + kcol + ((ln >> 4) << 1);
  v2f r;
  r.x = p[0];
  r.y = p[1];
  return r;
}

// B fragment where B[K][N] = src[K*stride + nofs + N]  (K is the row index):
//   elem(lane) = src[(kcol + (lane/16)*2 + {0,1})*stride + nofs + lane%16]
__device__ __forceinline__ v2f frag_cm(const float* __restrict__ src, int stride,
                                       int kcol, int nofs) {
  const int ln = threadIdx.x & 31;
  const float* p = src + (size_t)(kcol + ((ln >> 4) << 1)) * stride + nofs + (ln & 15);
  v2f r;
  r.x = p[0];
  r.y = p[stride];
  return r;
}

// ---------------- wave32 reductions ----------------

__device__ __forceinline__ float wave_sum(float v) {
  for (int m = 16; m >= 1; m >>= 1) v += __shfl_xor(v, m, 32);
  return v;
}
__device__ __forceinline__ float wave_max(float v) {
  for (int m = 16; m >= 1; m >>= 1) v = fmaxf(v, __shfl_xor(v, m, 32));
  return v;
}
// argmax; ties -> lowest index (matches lax.top_k ordering semantics)
__device__ __forceinline__ void wave_argmax(float& v, int& i) {
  for (int m = 16; m >= 1; m >>= 1) {
    float ov = __shfl_xor(v, m, 32);
    int oi = __shfl_xor(i, m, 32);
    if (ov > v || (ov == v && oi < i)) { v = ov; i = oi; }
  }
}

// ---------------- K0: pack valid bits ----------------
// vbits[b*64 + w] bit i  <=>  valid[b][w*32+i]
__global__ void k_pack_valid(const unsigned char* __restrict__ valid,
                             unsigned* __restrict__ vbits,
                             unsigned* __restrict__ hasvalid) {
  const int t = threadIdx.x;  // 256 threads = B_*64 words
  const int b = t >> 6, w = t & 63;
  unsigned bits = 0;
  for (int i = 0; i < 32; i++)
    bits |= (valid[b * N_ + w * 32 + i] ? 1u : 0u) << i;
  vbits[b * 64 + w] = bits;
  __syncthreads();
  if (w == 0) {
    unsigned any = 0;
    for (int j = 0; j < 64; j++) any |= vbits[b * 64 + j];
    hasvalid[b] = any ? 1u : 0u;
  }
}

// ---------------- K1: row L2-normalize ----------------
__global__ void k_normalize(const float* __restrict__ x, float* __restrict__ normed) {
  const int row = blockIdx.x;  // b*N + n, 256 threads = D_
  const float v = x[(size_t)row * D_ + threadIdx.x];
  __shared__ float red[8];
  __shared__ float s_r;
  float w = wave_sum(v * v);
  if ((threadIdx.x & 31) == 0) red[threadIdx.x >> 5] = w;
  __syncthreads();
  if (threadIdx.x == 0) {
    float t = 0;
    for (int i = 0; i < 8; i++) t += red[i];
    s_r = 1.0f / fmaxf(sqrtf(t), 1e-12f);
  }
  __syncthreads();
  normed[(size_t)row * D_ + threadIdx.x] = v * s_r;
}

// ---------------- K2: cosine sim (WMMA) + iterative top-40 + mask build ------
__global__ void __launch_bounds__(256) k_sim_topk(
    const float* __restrict__ normed, const unsigned* __restrict__ vbits,
    const unsigned* __restrict__ hasvalid, unsigned* __restrict__ amask) {
  extern __shared__ float sim[];  // [16][SIMLD]
  const int b = blockIdx.x >> 7;
  const int qbase = (blockIdx.x & 127) * 16;
  const int wv = threadIdx.x >> 5, ln = threadIdx.x & 31;
  const float* Nb = normed + (size_t)b * N_ * D_;
  const float* Aq = Nb + (size_t)qbase * D_;

  // 16 x 2048 similarity strip via fp32 WMMA
  for (int kt = wv; kt < N_ / 16; kt += 8) {
    const float* Bk = Nb + (size_t)kt * 16 * D_;
    v8f c = {};
    for (int ks = 0; ks < D_ / 4; ks++)
      c = wmma_f32(frag_rm(Aq, D_, ks * 4), frag_rm(Bk, D_, ks * 4), c);
    const int col = kt * 16 + (ln & 15);
    const int rb = (ln >> 4) * 8;
#pragma unroll
    for (int r = 0; r < 8; r++) sim[(rb + r) * SIMLD + col] = c[r];
  }
  __syncthreads();

  // top-40 per row; wave wv owns rows 2*wv and 2*wv+1 (full row in 64 VGPRs)
  for (int rr = 0; rr < 2; rr++) {
    const int row = wv * 2 + rr;
    const int q = qbase + row;
    const bool qvalid = (vbits[b * 64 + (q >> 5)] >> (q & 31)) & 1;
    unsigned m0 = 0, m1 = 0;  // this lane owns mask words 2*ln and 2*ln+1
    if (qvalid) {
      float rv[64];  // rv[i] = sim[row][i*32 + ln] (masked)
#pragma unroll
      for (int i = 0; i < 64; i++) {
        const unsigned vb = vbits[b * 64 + i];  // scalar (uniform) load
        const float s = sim[row * SIMLD + i * 32 + ln];
        rv[i] = ((vb >> ln) & 1) ? s : -INFINITY;
      }
      for (int t = 0; t < TOPK_; t++) {
        float bv = -INFINITY;
        int bi = N_;
#pragma unroll
        for (int i = 0; i < 64; i++) {
          const int k = i * 32 + ln;
          if (rv[i] > bv || (rv[i] == bv && k < bi)) { bv = rv[i]; bi = k; }
        }
        wave_argmax(bv, bi);  // all lanes get winning global key index bi
        if ((bi >> 6) == ln) {
          if ((bi >> 5) & 1) m1 |= 1u << (bi & 31);
          else               m0 |= 1u << (bi & 31);
        }
        const int clr = ((bi & 31) == ln) ? (bi >> 5) : -1;
#pragma unroll
        for (int i = 0; i < 64; i++)
          if (i == clr) rv[i] = -INFINITY;
      }
      // diagonal (q is valid here), then restrict to valid keys
      if ((q >> 6) == ln) {
        if ((q >> 5) & 1) m1 |= 1u << (q & 31);
        else              m0 |= 1u << (q & 31);
      }
      m0 &= vbits[b * 64 + 2 * ln];
      m1 &= vbits[b * 64 + 2 * ln + 1];
    } else {
      const unsigned hv = hasvalid[b];
      m0 = hv ? vbits[b * 64 + 2 * ln] : 0xFFFFFFFFu;
      m1 = hv ? vbits[b * 64 + 2 * ln + 1] : 0xFFFFFFFFu;
    }
    unsigned* outw = amask + ((size_t)(b * N_ + q)) * 64;
    outw[2 * ln] = m0;
    outw[2 * ln + 1] = m1;
  }
}

// ---------------- K3: QKV projection (WMMA) ----------------
__global__ void k_qkv(const float* __restrict__ x, const float* __restrict__ w,
                      const float* __restrict__ bias, float* __restrict__ qkv) {
  int id = blockIdx.x;  // B_*128*48 blocks, 32 threads (1 wave)
  const int jt = id % 48; id /= 48;
  const int nt = id % 128;
  const int b = id / 128;
  const float* A = x + (size_t)(b * N_ + nt * 16) * D_;
  const float* Bw = w + (size_t)(jt * 16) * D_;  // W[j][d] row-major
  v8f c = {};
  for (int ks = 0; ks < D_ / 4; ks++)
    c = wmma_f32(frag_rm(A, D_, ks * 4), frag_rm(Bw, D_, ks * 4), c);
  const int ln = threadIdx.x & 31;
  const int col = jt * 16 + (ln & 15);
  const float bb = bias[col];
  const int rb = (ln >> 4) * 8;
#pragma unroll
  for (int r = 0; r < 8; r++)
    qkv[(size_t)(b * N_ + nt * 16 + rb + r) * (3 * D_) + col] = c[r] + bb;
}

// ---------------- K4: masked attention for one (b, h, 16-query) strip --------
__global__ void __launch_bounds__(256) k_attn(const float* __restrict__ qkv,
                                              const unsigned* __restrict__ amask,
                                              float* __restrict__ o) {
  extern __shared__ float sc[];            // [16][SIMLD] scores/attn
  float* part = sc + 16 * SIMLD;           // [8][16][32] per-wave partials
  int id = blockIdx.x;
  const int qbase = (id & 127) * 16; id >>= 7;
  const int h = id & 7;
  const int b = id >> 3;
  const int wv = threadIdx.x >> 5, ln = threadIdx.x & 31;
  const float scale = 0.17677669529663687f;  // 1/sqrt(32)

  const float* Qb = qkv + (size_t)(b * N_ + qbase) * (3 * D_) + h * HD_;
  for (int kt = wv; kt < N_ / 16; kt += 8) {
    const float* Kb = qkv + (size_t)(b * N_ + kt * 16) * (3 * D_) + D_ + h * HD_;
    v8f c = {};
    for (int ks = 0; ks < HD_ / 4; ks++)
      c = wmma_f32(frag_rm(Qb, 3 * D_, ks * 4), frag_rm(Kb, 3 * D_, ks * 4), c);
    const int col = kt * 16 + (ln & 15);
    const int wofs = col >> 5, bit = col & 31;
    const int rb = (ln >> 4) * 8;
#pragma unroll
    for (int r = 0; r < 8; r++) {
      const int qrow = qbase + rb + r;
      const unsigned aw = amask[(size_t)(b * N_ + qrow) * 64 + wofs];
      sc[(rb + r) * SIMLD + col] = ((aw >> bit) & 1) ? c[r] * scale : -INFINITY;
    }
  }
  __syncthreads();

  // exact softmax; wave wv owns rows 2*wv, 2*wv+1
  for (int rr = 0; rr < 2; rr++) {
    const int row = wv * 2 + rr;
    float m = -INFINITY;
    for (int k = ln; k < N_; k += 32) m = fmaxf(m, sc[row * SIMLD + k]);
    m = wave_max(m);
    float ssum = 0.0f;
    for (int k = ln; k < N_; k += 32) {
      const float e = expf(sc[row * SIMLD + k] - m);
      sc[row * SIMLD + k] = e;
      ssum += e;
    }
    ssum = wave_sum(ssum);
    const float inv = 1.0f / ssum;
    for (int k = ln; k < N_; k += 32) sc[row * SIMLD + k] *= inv;
  }
  __syncthreads();

  // attn @ V : wave wv covers keys [wv*256, wv*256+256)
  const float* Vb = qkv + (size_t)b * N_ * (3 * D_) + 2 * D_ + h * HD_;
  v8f c0 = {}, c1 = {};
  for (int ks = 0; ks < 64; ks++) {
    const int kk = wv * 256 + ks * 4;
    const v2f a = frag_rm(sc, SIMLD, kk);          // A = attn rows from LDS
    c0 = wmma_f32(a, frag_cm(Vb, 3 * D_, kk, 0), c0);
    c1 = wmma_f32(a, frag_cm(Vb, 3 * D_, kk, 16), c1);
  }
  {
    const int rb = (ln >> 4) * 8, nloc = ln & 15;
#pragma unroll
    for (int r = 0; r < 8; r++) {
      part[(wv * 16 + rb + r) * 32 + nloc] = c0[r];
      part[(wv * 16 + rb + r) * 32 + 16 + nloc] = c1[r];
    }
  }
  __syncthreads();
  for (int t = threadIdx.x; t < 512; t += 256) {
    const int row = t >> 5, col = t & 31;
    float s = 0.0f;
    for (int w = 0; w < 8; w++) s += part[(w * 16 + row) * 32 + col];
    o[(size_t)(b * N_ + qbase + row) * D_ + h * HD_ + col] = s;
  }
}

// ---------------- K5: out-proj (WMMA) + residual + LayerNorm ----------------
__global__ void __launch_bounds__(256) k_out_ln(
    const float* __restrict__ o, const float* __restrict__ x,
    const float* __restrict__ ow, const float* __restrict__ ob,
    const float* __restrict__ gamma, const float* __restrict__ beta,
    float* __restrict__ out) {
  __shared__ float y[16 * 260];
  const int b = blockIdx.x >> 7;
  const int nbase = (blockIdx.x & 127) * 16;
  const int wv = threadIdx.x >> 5, ln = threadIdx.x & 31;
  const float* A = o + (size_t)(b * N_ + nbase) * D_;
  for (int jt = wv * 2; jt < wv * 2 + 2; jt++) {
    const float* Bw = ow + (size_t)(jt * 16) * D_;
    v8f c = {};
    for (int ks = 0; ks < D_ / 4; ks++)
      c = wmma_f32(frag_rm(A, D_, ks * 4), frag_rm(Bw, D_, ks * 4), c);
    const int col = jt * 16 + (ln & 15);
    const float bb = ob[col];
    const int rb = (ln >> 4) * 8;
#pragma unroll
    for (int r = 0; r < 8; r++) {
      const int row = rb + r;
      y[row * 260 + col] =
          c[r] + bb + x[(size_t)(b * N_ + nbase + row) * D_ + col];
    }
  }
  __syncthreads();
  for (int rr = 0; rr < 2; rr++) {
    const int row = wv * 2 + rr;
    float s = 0.0f, sq = 0.0f;
    for (int j = ln; j < D_; j += 32) {
      const float v = y[row * 260 + j];
      s += v;
      sq += v * v;
    }
    s = wave_sum(s);
    sq = wave_sum(sq);
    const float mu = s * (1.0f / D_);
    const float var = sq * (1.0f / D_) - mu * mu;  // biased, matches x.var
    const float inv = rsqrtf(var + LN_EPS);
    for (int j = ln; j < D_; j += 32) {
      const float v = (y[row * 260 + j] - mu) * inv;
      out[(size_t)(b * N_ + nbase + row) * D_ + j] = v * gamma[j] + beta[j];
    }
  }
}

// ---------------- launcher ----------------
extern "C" void kernel_launch(void* const* d_in, const int* in_sizes, int n_in,
                              void* d_out, int out_size, void* d_ws, size_t ws_size,
                              hipStream_t stream) {
  const float* x = (const float*)d_in[0];
  const unsigned char* valid = (const unsigned char*)d_in[1];  // jax bool = 1B
  const float* in_w = (const float*)d_in[2];
  const float* in_b = (const float*)d_in[3];
  const float* out_w = (const float*)d_in[4];
  const float* out_b = (const float*)d_in[5];
  const float* gamma = (const float*)d_in[6];
  const float* beta = (const float*)d_in[7];
  float* out = (float*)d_out;

  // workspace carve-up (~42 MB total)
  char* w = (char*)d_ws;
  float* normed = (float*)w;  w += (size_t)B_ * N_ * D_ * sizeof(float);       // 8 MB
  float* qkv    = (float*)w;  w += (size_t)B_ * N_ * 3 * D_ * sizeof(float);   // 24 MB
  float* o      = (float*)w;  w += (size_t)B_ * N_ * D_ * sizeof(float);       // 8 MB
  unsigned* vbits = (unsigned*)w;  w += (size_t)B_ * 64 * sizeof(unsigned);
  unsigned* hasv  = (unsigned*)w;  w += (size_t)B_ * sizeof(unsigned) * 4;     // keep 16B align
  unsigned* amask = (unsigned*)w;                                              // 2 MB

  k_pack_valid<<<1, 256, 0, stream>>>(valid, vbits, hasv);
  k_normalize<<<B_ * N_, 256, 0, stream>>>(x, normed);

  const size_t simBytes = (size_t)16 * SIMLD * sizeof(float);                  // ~128 KB
  k_sim_topk<<<B_ * (N_ / 16), 256, simBytes, stream>>>(normed, vbits, hasv, amask);

  k_qkv<<<B_ * (N_ / 16) * 48, 32, 0, stream>>>(x, in_w, in_b, qkv);

  const size_t attnBytes = simBytes + (size_t)8 * 16 * 32 * sizeof(float);     // ~144 KB
  k_attn<<<B_ * H_ * (N_ / 16), 256, attnBytes, stream>>>(qkv, amask, o);

  k_out_ln<<<B_ * (N_ / 16), 256, 0, stream>>>(o, x, out_w, out_b, gamma, beta, out);
}